// TokenEmbedding_81166291960021
// MI455X (gfx1250) — compile-verified
//
#include <hip/hip_runtime.h>

typedef __attribute__((ext_vector_type(2))) float v2f;
typedef __attribute__((ext_vector_type(8))) float v8f;

#define C_IN_    7
#define D_MODEL_ 512
#define KERNELS_ 73          // D_MODEL / C_IN
#define SLEN_    4096
#define WARM_    15          // M*TAO warm-up zeros
#define KK_      18          // (M+1)*3 contraction length
#define NTILE_   35          // 7 groups x 5 subtiles of 16 output channels
#define TT_      4           // 4 x 16 = 64 timesteps per block
#define XWIN_    (16 * TT_ + 17)            // 81 staged timesteps
#define BF_ELEMS (NTILE_ * 5 * 32 * 2)      // pre-packed B fragments (floats)

// ---------------------------------------------------------------------------
// Prep kernel (once per launch, 35 blocks x 32 threads): pre-pack WMMA
// B-matrix fragments (exact V_WMMA_F32_16X16X4_F32 lane/VGPR layout) and the
// merged bias vector into workspace. ~46 KB total -> stays L2-resident.
// ---------------------------------------------------------------------------
__global__ __launch_bounds__(32)
void lagconv_prep_kernel(const float* __restrict__ group_w,  // [511,6,3]
                         const float* __restrict__ group_b,  // [511]
                         const float* __restrict__ left_w,   // [1,6,3]
                         const float* __restrict__ left_b,   // [1]
                         float* __restrict__ bfrag,          // [35][5][32][2]
                         float* __restrict__ biasm)          // [512]
{
    const int tile = blockIdx.x;
    const int lane = threadIdx.x;
    const int g = tile / 5, j = tile % 5;
    const int o0 = g * KERNELS_ + 16 * j;
    int wlim = (g < 6 ? KERNELS_ : KERNELS_ + 1) - 16 * j;
    if (wlim > 16) wlim = 16;
    const int half = lane >> 4, l16 = lane & 15;
    const int o = o0 + l16;

#pragma unroll
    for (int s = 0; s < 5; ++s) {
        float f0 = 0.0f, f1 = 0.0f;
        const int kk = 4 * s + 2 * half;
        if (l16 < wlim) {
            if (kk < KK_)     f0 = (o < 511) ? group_w[o * KK_ + kk]     : left_w[kk];
            if (kk + 1 < KK_) f1 = (o < 511) ? group_w[o * KK_ + kk + 1] : left_w[kk + 1];
        }
        float* p = bfrag + ((size_t)(tile * 5 + s) * 32 + lane) * 2;
        p[0] = f0;
        p[1] = f1;
    }
    if (tile == 0)
        for (int i = lane; i < D_MODEL_; i += 32)
            biasm[i] = (i < 511) ? group_b[i] : left_b[0];
}

// ---------------------------------------------------------------------------
// Main kernel: 224 threads = 7 waves; wave w owns channel group g = w.
// Block covers 64 timesteps (4 sixteen-row t-subtiles). A-fragments for all
// 4 subtiles are built once per wave; B-fragments/bias load once per output
// subtile and feed 4 accumulations -> 100 WMMAs per wave.
// ---------------------------------------------------------------------------
__global__ __launch_bounds__(224)
void lagconv_wmma_kernel(const float* __restrict__ x,       // [B, S, 7]
                         const float* __restrict__ bfrag,   // [35][5][32][2]
                         const float* __restrict__ biasm,   // [512]
                         float* __restrict__ out)           // [B, S, 512]
{
    __shared__ float xs[XWIN_ * C_IN_];  // flat window: times t0-16 .. t0+64

    const int t0  = blockIdx.x * (16 * TT_);
    const int b   = blockIdx.y;
    const int tid = threadIdx.x;

    // Flat circular copy: wrap is a contiguous +/- SLEN*7 shift in flat space
    {
        const int fbase = (t0 - 16) * C_IN_;
        const size_t xb = (size_t)b * (SLEN_ * C_IN_);
        for (int i = tid; i < XWIN_ * C_IN_; i += 224) {
            int f = fbase + i;
            if (f < 0)                    f += SLEN_ * C_IN_;
            else if (f >= SLEN_ * C_IN_)  f -= SLEN_ * C_IN_;
            xs[i] = x[xb + f];
        }
    }
    __syncthreads();

    const int g    = tid >> 5;       // wave id == channel group
    const int lane = tid & 31;
    const int half = lane >> 4;
    const int l16  = lane & 15;

    // ---- build A fragments for all 4 t-subtiles (depend only on g, t0) ----
    v2f A[TT_][5];
#pragma unroll
    for (int tt = 0; tt < TT_; ++tt) {
#pragma unroll
        for (int s = 0; s < 5; ++s) {
#pragma unroll
            for (int h = 0; h < 2; ++h) {
                // compile-time (m,k) decomposition for both lane halves
                const int kk0 = 4 * s + h;          // half == 0
                const int kk1 = 4 * s + 2 + h;      // half == 1
                const int k0  = kk0 % 3,  m0 = kk0 / 3;
                const int k1  = kk1 % 3,  m1 = kk1 / 3;
                const bool ok1 = (kk1 < KK_);       // false only for s == 4
                const int dq0 = k0 - 1 - 3 * m0 + 16;
                const int dq1 = ok1 ? (k1 - 1 - 3 * m1 + 16) : 0;

                const int  kc    = half ? k1  : k0;   // cndmask of constants
                const int  dq    = half ? dq1 : dq0;
                const bool valid = half ? ok1 : true;

                const int   u  = (t0 + 16 * tt + l16 + kc - 1) & (SLEN_ - 1);
                const float xv = xs[(16 * tt + l16 + dq) * C_IN_ + g];
                const float av = (valid && u >= WARM_) ? xv : 0.0f;
                if (h == 0) A[tt][s].x = av; else A[tt][s].y = av;
            }
        }
    }

    // ---- 5 output subtiles of this group ----
#pragma unroll
    for (int j = 0; j < 5; ++j) {
        const int tile = g * 5 + j;
        const int o0   = g * KERNELS_ + 16 * j;
        int wlim = (g < 6 ? KERNELS_ : KERNELS_ + 1) - 16 * j;
        if (wlim > 16) wlim = 16;

        v2f Bf[5];
#pragma unroll
        for (int s = 0; s < 5; ++s)
            Bf[s] = *(const v2f*)(bfrag +
                                  ((size_t)(tile * 5 + s) * 32 + lane) * 2);
        const float bn = (l16 < wlim) ? biasm[o0 + l16] : 0.0f;

#pragma unroll
        for (int tt = 0; tt < TT_; ++tt) {
            v8f acc = {};
#pragma unroll
            for (int s = 0; s < 5; ++s)
                acc = __builtin_amdgcn_wmma_f32_16x16x4_f32(
                          false, A[tt][s], false, Bf[s],
                          (short)0, acc, false, false);

            if (l16 < wlim) {
                const size_t base =
                    ((size_t)b * SLEN_ + t0 + 16 * tt + 8 * half) * D_MODEL_ +
                    o0 + l16;
#pragma unroll
                for (int rr = 0; rr < 8; ++rr)
                    out[base + (size_t)rr * D_MODEL_] = acc[rr] + bn;
            }
        }
    }
}

extern "C" void kernel_launch(void* const* d_in, const int* in_sizes, int n_in,
                              void* d_out, int out_size, void* d_ws, size_t ws_size,
                              hipStream_t stream) {
    (void)in_sizes; (void)n_in; (void)out_size; (void)ws_size;
    const float* x       = (const float*)d_in[0];
    const float* group_w = (const float*)d_in[1];
    const float* group_b = (const float*)d_in[2];
    const float* left_w  = (const float*)d_in[3];
    const float* left_b  = (const float*)d_in[4];
    float* out = (float*)d_out;

    float* bfrag = (float*)d_ws;            // 44.8 KB packed B fragments
    float* biasm = bfrag + BF_ELEMS;        // + 2 KB merged bias

    lagconv_prep_kernel<<<NTILE_, 32, 0, stream>>>(group_w, group_b,
                                                   left_w, left_b,
                                                   bfrag, biasm);
    dim3 grid(SLEN_ / (16 * TT_), 32);      // 64 time blocks x 32 batches
    lagconv_wmma_kernel<<<grid, 224, 0, stream>>>(x, bfrag, biasm, out);
}